// gcnlayer2_77163382440859
// MI455X (gfx1250) — compile-verified
//
#include <hip/hip_runtime.h>
#include <stdint.h>

#define NNODES 65536
#define DFEAT  64

// ---------------------------------------------------------------------------
// Zero-init of d_out (harness poisons it with 0xAA before timing).
// ---------------------------------------------------------------------------
__global__ void __launch_bounds__(256) gcn_zero_kernel(float4* __restrict__ out, int n4) {
    int i = blockIdx.x * blockDim.x + threadIdx.x;
    if (i < n4) out[i] = make_float4(0.f, 0.f, 0.f, 0.f);
}

// Low 32 bits of a generic pointer into __shared__ == LDS byte offset.
__device__ __forceinline__ uint32_t lds_off(const void* p) {
    return (uint32_t)(uintptr_t)p;
}

// ---------------------------------------------------------------------------
// COO SpMM scatter: out[rows[e]] += vals[e] * x[cols[e]]   (D = 64 f32)
//
// Block = 256 threads = 8 wave32. Each wave handles 32 consecutive edges:
//   - stages rows/cols/vals for its 32 edges into LDS via async global->LDS
//     copies (CDNA5 ASYNCcnt path), waits with s_wait_asynccnt (wave-private,
//     so no workgroup barrier required),
//   - then 16 passes: lanes 0-15 take edge 2p, lanes 16-31 take edge 2p+1,
//     each lane owns a float4 feature chunk (16 lanes x 4 = 64 features),
//   - gather x row chunk (coalesced 256B/edge), scale, 4x f32 atomic adds.
// ---------------------------------------------------------------------------
__global__ void __launch_bounds__(256) gcn_spmm_scatter_kernel(
    const int*   __restrict__ rows,
    const int*   __restrict__ cols,
    const float* __restrict__ vals,
    const float* __restrict__ x,
    float*       __restrict__ out,
    int nEdges)
{
    __shared__ int   s_rows[256];
    __shared__ int   s_cols[256];
    __shared__ float s_vals[256];

    const int tid      = threadIdx.x;
    const int lane     = tid & 31;
    const int wave     = tid >> 5;
    const int waveBase = blockIdx.x * 256 + wave * 32;
    const int e        = waveBase + lane;

    // ---- async stage of this wave's 32 edges into LDS -----------------------
    if (e < nEdges) {
        const uint32_t lr  = lds_off(&s_rows[wave * 32 + lane]);
        const uint32_t lc  = lds_off(&s_cols[wave * 32 + lane]);
        const uint32_t lv  = lds_off(&s_vals[wave * 32 + lane]);
        const uint32_t off = (uint32_t)e * 4u;   // byte offset into each stream
        asm volatile(
            "global_load_async_to_lds_b32 %0, %3, %4\n\t"
            "global_load_async_to_lds_b32 %1, %3, %5\n\t"
            "global_load_async_to_lds_b32 %2, %3, %6"
            :
            : "v"(lr), "v"(lc), "v"(lv), "v"(off),
              "s"(rows), "s"(cols), "s"(vals)
            : "memory");
    }
    asm volatile("s_wait_asynccnt 0" ::: "memory");
    // Wave-private staging: the issuing wave is the only consumer -> no barrier.

    const int chunk = (lane & 15) * 4;  // feature sub-block: 4 floats
    const int sub   = lane >> 4;        // which of the 2 edges this half-wave owns

    #pragma unroll
    for (int p = 0; p < 16; ++p) {
        const int eloc = p * 2 + sub;           // 0..31 within the wave's tile
        if (waveBase + eloc >= nEdges) continue;

        const int   row = s_rows[wave * 32 + eloc];
        const int   col = s_cols[wave * 32 + eloc];
        const float val = s_vals[wave * 32 + eloc];

        const float4 xv =
            *reinterpret_cast<const float4*>(x + (size_t)col * DFEAT + chunk);

        float* op = out + (size_t)row * DFEAT + chunk;
        __hip_atomic_fetch_add(op + 0, val * xv.x, __ATOMIC_RELAXED, __HIP_MEMORY_SCOPE_AGENT);
        __hip_atomic_fetch_add(op + 1, val * xv.y, __ATOMIC_RELAXED, __HIP_MEMORY_SCOPE_AGENT);
        __hip_atomic_fetch_add(op + 2, val * xv.z, __ATOMIC_RELAXED, __HIP_MEMORY_SCOPE_AGENT);
        __hip_atomic_fetch_add(op + 3, val * xv.w, __ATOMIC_RELAXED, __HIP_MEMORY_SCOPE_AGENT);
    }
}

extern "C" void kernel_launch(void* const* d_in, const int* in_sizes, int n_in,
                              void* d_out, int out_size, void* d_ws, size_t ws_size,
                              hipStream_t stream) {
    const float* x1      = (const float*)d_in[0];
    const float* x2      = (const float*)d_in[1];
    const int*   a1_rows = (const int*)  d_in[2];
    const int*   a1_cols = (const int*)  d_in[3];
    const float* a1_vals = (const float*)d_in[4];
    const int*   a2_rows = (const int*)  d_in[5];
    const int*   a2_cols = (const int*)  d_in[6];
    const float* a2_vals = (const float*)d_in[7];

    float* out1 = (float*)d_out;
    float* out2 = out1 + (size_t)NNODES * DFEAT;

    const int E1 = in_sizes[2];
    const int E2 = in_sizes[5];

    // 1) zero the outputs (f32 accumulation target)
    const int n4 = out_size / 4;
    gcn_zero_kernel<<<(n4 + 255) / 256, 256, 0, stream>>>((float4*)d_out, n4);

    // 2) two independent scatter SpMMs (same stream -> ordered after the zero)
    gcn_spmm_scatter_kernel<<<(E1 + 255) / 256, 256, 0, stream>>>(
        a1_rows, a1_cols, a1_vals, x1, out1, E1);
    gcn_spmm_scatter_kernel<<<(E2 + 255) / 256, 256, 0, stream>>>(
        a2_rows, a2_cols, a2_vals, x2, out2, E2);
}